// FocalLoss_11527692222863
// MI455X (gfx1250) — compile-verified
//
#include <hip/hip_runtime.h>
#include <stdint.h>

typedef __attribute__((ext_vector_type(2))) float v2f;
typedef __attribute__((ext_vector_type(8))) float v8f;

#define BATCH      8
#define ANCH       60000
#define NA         (BATCH * ANCH)        // 480000 anchors
#define NPAIRS     (NA / 2)              // 240000 anchor pairs
#define CDIM       150
#define TAB_ROWS   21                    // G+1
#define TAB_WPR    5                     // ceil(150/32)
#define TAB_WORDS  (BATCH * TAB_ROWS * TAB_WPR)  // 840

#define NB_CLS     1500
#define NB_REG     512

// workspace layout (bytes)
#define WS_TAB     0
#define WS_CLS     4096
#define WS_REG     (WS_CLS + 8192)
#define WS_POS     (WS_REG + 4096)

// ---------------------------------------------------------------------------
// Kernel 1: build per-(batch,row) class-presence bitmask table (840 u32).
// row 0 == zeros (padding row for labels_bin <= 0), rows 1..20 from groups.
// ---------------------------------------------------------------------------
__global__ __launch_bounds__(1024)
void build_table_kernel(const int* __restrict__ gt_labels, unsigned* __restrict__ tabG) {
    __shared__ unsigned st[TAB_WORDS];
    const int tid = threadIdx.x;
    for (int i = tid; i < TAB_WORDS; i += blockDim.x) st[i] = 0u;
    __syncthreads();
    // gt_labels flat: (B=8, G=20, NLT=5, T=5) -> 4000 entries
    for (int j = tid; j < 4000; j += blockDim.x) {
        int head = (j / 5) % 5;
        int g    = (j / 25) % 20;
        int b    =  j / 500;
        int lab  = gt_labels[j];
        if (lab >= 0) {
            int off;   // confidence-channel offset of each head
            switch (head) {
                case 0: off = 1;   break;
                case 1: off = 11;  break;
                case 2: off = 30;  break;
                case 3: off = 69;  break;
                default: off = 138; break;
            }
            int c   = off + lab;                      // 1..149
            int row = b * TAB_ROWS + (g + 1);
            atomicOr(&st[row * TAB_WPR + (c >> 5)], 1u << (c & 31));
        }
    }
    __syncthreads();
    for (int i = tid; i < TAB_WORDS; i += blockDim.x) tabG[i] = st[i];
}

// ---------------------------------------------------------------------------
// focal term: -log(q)*alpha*(1-q)^2*mask with q = l ? p : 1-p, p = sigmoid(x)
// q = 1/(1+t), t = exp(-(l ? x : -x)); clamp p in [1e-7,1-1e-7] <=> clamp t.
// ---------------------------------------------------------------------------
__device__ __forceinline__ float focal_term(float x, int lab, float m) {
    float xs = lab ? x : -x;
    float t  = __builtin_amdgcn_exp2f(-1.44269504f * xs);
    t = fminf(fmaxf(t, 1.0000001e-7f), 9999999.0f);
    float u  = 1.0f + t;
    float r  = __builtin_amdgcn_rcpf(u);    // 1/u
    float lg = __builtin_amdgcn_logf(u);    // log2(u);  -log(q) = lg * ln2
    float oq = t * r;                       // 1 - q
    float coef = m * (lab ? 0.1732868f : 0.5198604f);  // alpha * ln2 * mask
    return lg * coef * (oq * oq);
}

// ---------------------------------------------------------------------------
// Kernel 2: classification focal loss, streaming 288 MB of confidence.
// 80-lane groups own one anchor PAIR (1200 B, float4-aligned); lanes 0..74
// each own 4 fixed confidence channels -> all index math hoisted out of loop.
// ---------------------------------------------------------------------------
__global__ __launch_bounds__(320)
void cls_kernel(const float* __restrict__ conf, const int* __restrict__ lbin,
                const unsigned* __restrict__ tabG, float* __restrict__ partial) {
    __shared__ unsigned tab[TAB_WORDS];
    __shared__ float sbuf[16];
    const int tid = threadIdx.x;
    for (int i = tid; i < TAB_WORDS; i += 320) tab[i] = tabG[i];
    __syncthreads();

    const int grp  = tid / 80;
    const int tsub = tid - grp * 80;
    const bool active = (tsub < 75);

    // per-thread compile-time-ish constants (4 channels of the pair's 300)
    int aoffj[4], wordj[4], bitj[4], isc0j[4];
#pragma unroll
    for (int j = 0; j < 4; ++j) {
        int e = tsub * 4 + j;                 // 0..299 within the pair
        int ao = (e >= CDIM) ? 1 : 0;
        int c  = e - CDIM * ao;               // channel 0..149
        aoffj[j] = ao;
        wordj[j] = c >> 5;
        bitj[j]  = c & 31;
        isc0j[j] = (c == 0) ? 1 : 0;
    }

    const float4* __restrict__ conf4 = (const float4*)conf;
    const int2*   __restrict__ lb2   = (const int2*)lbin;
    const int strideP = gridDim.x * 4;

    float acc = 0.0f;
    for (int pair = blockIdx.x * 4 + grp; pair < NPAIRS; pair += strideP) {
        if (active) {
            __builtin_prefetch(&conf4[(size_t)(pair + strideP) * 75 + tsub], 0, 0);
            int2 lbs = lb2[pair];
            int bb   = (int)((unsigned)pair / 30000u);           // batch index
            int idx0 = lbs.x > 0 ? lbs.x : 0;
            int idx1 = lbs.y > 0 ? lbs.y : 0;
            int row0 = (bb * TAB_ROWS + idx0) * TAB_WPR;
            int row1 = (bb * TAB_ROWS + idx1) * TAB_WPR;
            float mask0 = lbs.x > -1 ? 1.0f : 0.0f;
            float mask1 = lbs.y > -1 ? 1.0f : 0.0f;
            int pos0 = lbs.x > 0, pos1 = lbs.y > 0;

            float4 x4 = conf4[(size_t)pair * 75 + tsub];
            const float* xs = (const float*)&x4;
#pragma unroll
            for (int j = 0; j < 4; ++j) {
                unsigned w = tab[(aoffj[j] ? row1 : row0) + wordj[j]];
                int lab  = isc0j[j] ? (aoffj[j] ? pos1 : pos0)
                                    : (int)((w >> bitj[j]) & 1u);
                float m  = aoffj[j] ? mask1 : mask0;
                acc += focal_term(xs[j], lab, m);
            }
        }
    }

    // block reduction (10 waves)
#pragma unroll
    for (int off = 16; off > 0; off >>= 1) acc += __shfl_xor(acc, off);
    if ((tid & 31) == 0) sbuf[tid >> 5] = acc;
    __syncthreads();
    if (tid == 0) {
        float s = 0.0f;
        for (int i = 0; i < 10; ++i) s += sbuf[i];
        partial[blockIdx.x] = s;
    }
}

// ---------------------------------------------------------------------------
// Kernel 3: smooth-L1 regression sum (pos anchors only) + positive count.
// ---------------------------------------------------------------------------
__global__ __launch_bounds__(256)
void reg_kernel(const float* __restrict__ pred, const float* __restrict__ gt,
                const int* __restrict__ lbin,
                float* __restrict__ partialR, float* __restrict__ partialP) {
    __shared__ float sbufS[8];
    __shared__ float sbufP[8];
    const int tid = threadIdx.x;
    const float4* __restrict__ p4 = (const float4*)pred;
    const float4* __restrict__ g4 = (const float4*)gt;

    float accS = 0.0f, accP = 0.0f;
    const int stride = gridDim.x * blockDim.x;
    for (int a = blockIdx.x * blockDim.x + tid; a < NA; a += stride) {
        int lb = lbin[a];
        if (lb > 0) {
            accP += 1.0f;
            float4 p = p4[a];
            float4 g = g4[a];
#pragma unroll
            for (int j = 0; j < 4; ++j) {
                float d = ((const float*)&p)[j] - ((const float*)&g)[j];
                float n = fabsf(d);
                accS += (n < (1.0f / 9.0f)) ? 4.5f * n * n
                                            : n - (1.0f / 18.0f);
            }
        }
    }
#pragma unroll
    for (int off = 16; off > 0; off >>= 1) {
        accS += __shfl_xor(accS, off);
        accP += __shfl_xor(accP, off);
    }
    if ((tid & 31) == 0) { sbufS[tid >> 5] = accS; sbufP[tid >> 5] = accP; }
    __syncthreads();
    if (tid == 0) {
        float s = 0.0f, p = 0.0f;
        for (int i = 0; i < 8; ++i) { s += sbufS[i]; p += sbufP[i]; }
        partialR[blockIdx.x] = s;
        partialP[blockIdx.x] = p;
    }
}

// ---------------------------------------------------------------------------
// Final reduction on the matrix pipe: v_wmma_f32_16x16x4_f32 with all-ones A
// is a 64-way adder tree (D[m][n] = sum_k B[k][n], accumulated through C).
// One wave, EXEC all ones as WMMA requires.
// ---------------------------------------------------------------------------
__device__ __forceinline__ float wmma_sum(const float* __restrict__ arr, int n) {
    const int lane = threadIdx.x & 31;
    v2f ones; ones.x = 1.0f; ones.y = 1.0f;
    v8f acc = {};
    for (int base = 0; base < n; base += 64) {
        v2f b; b.x = 0.0f; b.y = 0.0f;
        int i0 = base + lane;
        int i1 = base + 32 + lane;
        if (i0 < n) b.x = arr[i0];
        if (i1 < n) b.y = arr[i1];
        acc = __builtin_amdgcn_wmma_f32_16x16x4_f32(
            /*neg_a=*/false, ones, /*neg_b=*/false, b,
            /*c_mod=*/(short)0, acc, /*reuse_a=*/false, /*reuse_b=*/false);
    }
    // acc[0] holds column-sums for N = lane%16 (rows identical): reduce 16 lanes
    float v = acc[0];
#pragma unroll
    for (int off = 1; off < 16; off <<= 1) v += __shfl_xor(v, off);
    return v;  // every lane now holds the total
}

__global__ __launch_bounds__(32)
void final_kernel(const float* __restrict__ pc, const float* __restrict__ pr,
                  const float* __restrict__ pp, float* __restrict__ out) {
    float Sc = wmma_sum(pc, NB_CLS);
    float Sr = wmma_sum(pr, NB_REG);
    float Sp = wmma_sum(pp, NB_REG);
    if (threadIdx.x == 0) {
        float np = fmaxf(1.0f, Sp);
        out[0] = Sr / (np * 4.0f);   // regression_loss
        out[1] = Sc / (np * 6.0f);   // cls_loss / 6
    }
}

// ---------------------------------------------------------------------------
extern "C" void kernel_launch(void* const* d_in, const int* in_sizes, int n_in,
                              void* d_out, int out_size, void* d_ws, size_t ws_size,
                              hipStream_t stream) {
    const float* conf  = (const float*)d_in[0];
    const float* pred  = (const float*)d_in[1];
    const float* gt    = (const float*)d_in[2];
    const int*   glab  = (const int*)d_in[3];
    // d_in[4] = counts (unused by the reference loss body)
    const int*   lbin  = (const int*)d_in[5];
    float* out = (float*)d_out;

    char* ws = (char*)d_ws;
    unsigned* tabG     = (unsigned*)(ws + WS_TAB);
    float*    partialC = (float*)(ws + WS_CLS);
    float*    partialR = (float*)(ws + WS_REG);
    float*    partialP = (float*)(ws + WS_POS);

    build_table_kernel<<<1, 1024, 0, stream>>>(glab, tabG);
    cls_kernel<<<NB_CLS, 320, 0, stream>>>(conf, lbin, tabG, partialC);
    reg_kernel<<<NB_REG, 256, 0, stream>>>(pred, gt, lbin, partialR, partialP);
    final_kernel<<<1, 32, 0, stream>>>(partialC, partialR, partialP, out);
}